// MemoryRetrieval_60550448939397
// MI455X (gfx1250) — compile-verified
//
#include <hip/hip_runtime.h>
#include <hip/hip_bf16.h>

typedef __attribute__((ext_vector_type(2))) float v2f;
typedef __attribute__((ext_vector_type(8))) float v8f;
typedef __attribute__((ext_vector_type(4))) int   v4i;

#define B_ROWS 4096
#define N_KEYS 32768
#define DIM    1024

#define BM 64
#define BN 256
#define KC 16
#define LDST 20   // padded LDS row stride (floats): 80B rows -> 16B aligned, conflict-free
#define NT (DIM / KC)

#if __has_builtin(__builtin_amdgcn_global_load_async_to_lds_b128)
#define HAVE_ASYNC_LDS 1
#define GAS __attribute__((address_space(1)))
#define LAS __attribute__((address_space(3)))
#endif

#if defined(HAVE_ASYNC_LDS)
// (int4 addrspace(1)*, int4 addrspace(3)*, imm offset, imm cpol)
__device__ __forceinline__ void async_cp16(const float* g, float* l) {
    __builtin_amdgcn_global_load_async_to_lds_b128(
        (GAS v4i*)const_cast<float*>(g), (LAS v4i*)l, 0, 0);
}
#endif

__device__ __forceinline__ void wait_async_lds() {
#if defined(HAVE_ASYNC_LDS)
#if __has_builtin(__builtin_amdgcn_s_wait_asynccnt)
    __builtin_amdgcn_s_wait_asynccnt(0);
#else
    asm volatile("s_wait_asynccnt 0x0" ::: "memory");
#endif
#endif
}

__device__ __forceinline__ unsigned order_f32(float x) {
    unsigned u = __float_as_uint(x);
    return (u & 0x80000000u) ? ~u : (u | 0x80000000u);
}

// ---------------- init per-row accumulators ----------------
__global__ void mr_init_kernel(float* __restrict__ sum, float* __restrict__ sumsq,
                               unsigned long long* __restrict__ packed) {
    int i = blockIdx.x * blockDim.x + threadIdx.x;
    if (i < B_ROWS) {
        sum[i] = 0.0f;
        sumsq[i] = 0.0f;
        packed[i] = 0ull;   // below ordered encoding of any real value
    }
}

// ---------------- inverse L2 norms per row ----------------
__global__ __launch_bounds__(256) void mr_rnorm_kernel(const float* __restrict__ x,
                                                       float* __restrict__ rn) {
    __shared__ float red[256];
    const int row = blockIdx.x;
    const float4 v = ((const float4*)(x + (size_t)row * DIM))[threadIdx.x];
    float t = v.x * v.x + v.y * v.y + v.z * v.z + v.w * v.w;
    red[threadIdx.x] = t;
    __syncthreads();
    for (int o = 128; o > 0; o >>= 1) {
        if (threadIdx.x < o) red[threadIdx.x] += red[threadIdx.x + o];
        __syncthreads();
    }
    if (threadIdx.x == 0) rn[row] = 1.0f / fmaxf(sqrtf(red[0]), 1e-12f);
}

// ---------------- main WMMA similarity + fused reduction ----------------
// grid: (N_KEYS/BN, B_ROWS/BM), block: 256 threads = 8 waves (wave32)
// wave (mw = w>>1, nw = w&1) owns rows [16*mw,+16) x cols [128*nw,+128)
__global__ __launch_bounds__(256) void mr_sim_kernel(
    const float* __restrict__ q, const float* __restrict__ k,
    const float* __restrict__ rnq, const float* __restrict__ rnk,
    float* __restrict__ sum, float* __restrict__ sumsq,
    unsigned long long* __restrict__ packed)
{
    __shared__ float As[2][BM * LDST];   //  2 x  5 KB
    __shared__ float Bs[2][BN * LDST];   //  2 x 20 KB

    const int tid  = threadIdx.x;
    const int wave = tid >> 5;
    const int lane = tid & 31;
    const int mw   = wave >> 1;
    const int nw   = wave & 1;
    const int m0   = blockIdx.y * BM;
    const int n0   = blockIdx.x * BN;

    v8f acc[8] = {};

    // global->LDS fill pattern: 256 threads, one B128 per (row, quarter-row)
    const int frow = tid >> 2;          // 0..63
    const int fcol = (tid & 3) * 4;     // 0,4,8,12
    const float* qrow = q + (size_t)(m0 + frow) * DIM + fcol;
    const float* krow = k + (size_t)(n0 + frow) * DIM + fcol;

    const int arow = mw * 16 + (lane & 15);
    const int koff = (lane >> 4) * 2;   // lanes 0-15: K=0,1 ; lanes 16-31: K=2,3

    // ---- prologue: fill buffer 0 ----
#if defined(HAVE_ASYNC_LDS)
    async_cp16(qrow, &As[0][frow * LDST + fcol]);
#pragma unroll
    for (int r = 0; r < 4; ++r)
        async_cp16(krow + (size_t)(64 * r) * DIM, &Bs[0][(frow + 64 * r) * LDST + fcol]);
    wait_async_lds();
#else
    {
        float4 ra = *(const float4*)qrow;
        float4 rb[4];
#pragma unroll
        for (int r = 0; r < 4; ++r) rb[r] = *(const float4*)(krow + (size_t)(64 * r) * DIM);
        *(float4*)&As[0][frow * LDST + fcol] = ra;
#pragma unroll
        for (int r = 0; r < 4; ++r) *(float4*)&Bs[0][(frow + 64 * r) * LDST + fcol] = rb[r];
    }
#endif
    __syncthreads();

    for (int t = 0; t < NT; ++t) {
        const int  cur      = t & 1;
        const int  nxt      = cur ^ 1;
        const bool has_next = (t + 1) < NT;
        const int  kc_next  = (t + 1) * KC;

#if defined(HAVE_ASYNC_LDS)
        if (has_next) {
            async_cp16(qrow + kc_next, &As[nxt][frow * LDST + fcol]);
#pragma unroll
            for (int r = 0; r < 4; ++r)
                async_cp16(krow + (size_t)(64 * r) * DIM + kc_next,
                           &Bs[nxt][(frow + 64 * r) * LDST + fcol]);
        }
#else
        float4 ra;
        float4 rb[4];
        if (has_next) {
            ra = *(const float4*)(qrow + kc_next);
#pragma unroll
            for (int r = 0; r < 4; ++r)
                rb[r] = *(const float4*)(krow + (size_t)(64 * r) * DIM + kc_next);
        }
#endif

        // ---- compute on buffer `cur` ----
        const float* Ac = As[cur];
        const float* Bc = Bs[cur];
#pragma unroll
        for (int kk = 0; kk < KC; kk += 4) {
            v2f a = *(const v2f*)&Ac[arow * LDST + kk + koff];
            v2f bfr[8];
#pragma unroll
            for (int j = 0; j < 8; ++j) {
                const int bcol = nw * 128 + j * 16 + (lane & 15);
                bfr[j] = *(const v2f*)&Bc[bcol * LDST + kk + koff];
            }
#pragma unroll
            for (int j = 0; j < 8; ++j)
                acc[j] = __builtin_amdgcn_wmma_f32_16x16x4_f32(
                    false, a, false, bfr[j], (short)0, acc[j], false, false);
        }

#if defined(HAVE_ASYNC_LDS)
        if (has_next) wait_async_lds();
#else
        if (has_next) {
            *(float4*)&As[nxt][frow * LDST + fcol] = ra;
#pragma unroll
            for (int r = 0; r < 4; ++r)
                *(float4*)&Bs[nxt][(frow + 64 * r) * LDST + fcol] = rb[r];
        }
#endif
        __syncthreads();
    }

    // ---- epilogue: scale by inverse norms, reduce per row ----
    const int colbase = n0 + nw * 128 + (lane & 15);
    float rk[8];
#pragma unroll
    for (int j = 0; j < 8; ++j) rk[j] = rnk[colbase + j * 16];

    const int rowhalf = (lane >> 4) * 8;
#pragma unroll
    for (int i = 0; i < 8; ++i) {
        const int grow = m0 + mw * 16 + rowhalf + i;
        const float rq = rnq[grow];
        float s = 0.0f, ss = 0.0f;
        float mx = -3.402823466e+38f;
        unsigned mi = 0u;
#pragma unroll
        for (int j = 0; j < 8; ++j) {
            const float v = acc[j][i] * rq * rk[j];
            s += v;
            ss += v * v;
            if (v > mx) { mx = v; mi = (unsigned)(colbase + j * 16); }
        }
        // butterfly over the 16 lanes that hold this row's columns
        for (int off = 8; off > 0; off >>= 1) {
            s  += __shfl_xor(s, off, 16);
            ss += __shfl_xor(ss, off, 16);
            const float    omx = __shfl_xor(mx, off, 16);
            const unsigned omi = (unsigned)__shfl_xor((int)mi, off, 16);
            if (omx > mx || (omx == mx && omi < mi)) { mx = omx; mi = omi; }
        }
        if ((lane & 15) == 0) {
            atomicAdd(&sum[grow], s);
            atomicAdd(&sumsq[grow], ss);
            const unsigned long long p =
                ((unsigned long long)order_f32(mx) << 32) | (unsigned long long)(~mi);
            atomicMax(&packed[grow], p);
        }
    }
}

// ---------------- variance finalize: mean over rows of unbiased var ----------------
__global__ __launch_bounds__(256) void mr_var_kernel(const float* __restrict__ sum,
                                                     const float* __restrict__ sumsq,
                                                     float* __restrict__ out_scalar) {
    __shared__ float red[256];
    float t = 0.0f;
    for (int r = threadIdx.x; r < B_ROWS; r += 256) {
        const float s = sum[r];
        const float mean = s * (1.0f / (float)N_KEYS);
        t += (sumsq[r] - s * mean) * (1.0f / (float)(N_KEYS - 1));
    }
    red[threadIdx.x] = t;
    __syncthreads();
    for (int o = 128; o > 0; o >>= 1) {
        if (threadIdx.x < o) red[threadIdx.x] += red[threadIdx.x + o];
        __syncthreads();
    }
    if (threadIdx.x == 0) *out_scalar = red[0] * (1.0f / (float)B_ROWS);
}

// ---------------- gather values[argmax] ----------------
__global__ __launch_bounds__(256) void mr_gather_kernel(
    const unsigned long long* __restrict__ packed,
    const float* __restrict__ values, float* __restrict__ out) {
    const int row = blockIdx.x;
    const unsigned idx = ~(unsigned)(packed[row] & 0xFFFFFFFFull);
    const float4* src = (const float4*)(values + (size_t)idx * DIM);
    float4* dst = (float4*)(out + (size_t)row * DIM);
    dst[threadIdx.x] = src[threadIdx.x];
}

extern "C" void kernel_launch(void* const* d_in, const int* in_sizes, int n_in,
                              void* d_out, int out_size, void* d_ws, size_t ws_size,
                              hipStream_t stream) {
    const float* q = (const float*)d_in[0];   // [4096,1024]
    const float* k = (const float*)d_in[1];   // [32768,1024]
    const float* v = (const float*)d_in[2];   // [32768,1024]
    float* out = (float*)d_out;               // [4096*1024] retrieved + 1 scalar

    // workspace layout (~208 KB)
    unsigned long long* packed = (unsigned long long*)d_ws;       // 4096 u64
    float* sum   = (float*)((char*)d_ws + B_ROWS * 8);            // 4096 f32
    float* sumsq = sum + B_ROWS;                                  // 4096 f32
    float* rnq   = sumsq + B_ROWS;                                // 4096 f32
    float* rnk   = rnq + B_ROWS;                                  // 32768 f32

    mr_init_kernel<<<(B_ROWS + 255) / 256, 256, 0, stream>>>(sum, sumsq, packed);
    mr_rnorm_kernel<<<B_ROWS, 256, 0, stream>>>(q, rnq);
    mr_rnorm_kernel<<<N_KEYS, 256, 0, stream>>>(k, rnk);

    dim3 grid(N_KEYS / BN, B_ROWS / BM);   // (128, 64)
    mr_sim_kernel<<<grid, 256, 0, stream>>>(q, k, rnq, rnk, sum, sumsq, packed);

    mr_var_kernel<<<1, 256, 0, stream>>>(sum, sumsq, out + (size_t)B_ROWS * DIM);
    mr_gather_kernel<<<B_ROWS, 256, 0, stream>>>(packed, v, out);
}